// MoEMLP_79611513799334
// MI455X (gfx1250) — compile-verified
//
#include <hip/hip_runtime.h>

// ---------------- constants ----------------
namespace {
constexpr int kH  = 1024;
constexpr int kE  = 8;
constexpr int kI  = 2816;
constexpr int kNG = 2 * kI;      // 5632 gate+up columns
constexpr int kT  = 8192;        // B*S tokens
constexpr int kMT = 64;          // tokens per workgroup tile
constexpr int kChunk  = 128;     // intermediate columns per chunk (8 waves * 16)
constexpr int kNChunk = kI / kChunk;  // 22
constexpr int kKSlice = 64;      // x K-slice staged in LDS
}

typedef __attribute__((ext_vector_type(2))) float v2f;
typedef __attribute__((ext_vector_type(8))) float v8f;

static __device__ __forceinline__ v8f wmma_f32_16x16x4(v2f a, v2f b, v8f c) {
  // D = A(16x4 f32) * B(4x16 f32) + C(16x16 f32)
  return __builtin_amdgcn_wmma_f32_16x16x4_f32(false, a, false, b, (short)0, c,
                                               false, false);
}

// silu(g)*u with v_rcp_f32 instead of the IEEE-division expansion
static __device__ __forceinline__ float silu_mul(float g, float u) {
  const float s = __builtin_amdgcn_rcpf(1.0f + __expf(-g));  // sigmoid(g)
  return g * s * u;
}

// ---------------- router: logits -> softmax -> top2 -> dense combine W ----
__global__ __launch_bounds__(256)
void router_kernel(const float* __restrict__ x, const float* __restrict__ gw,
                   float* __restrict__ Wt, float* __restrict__ sums) {
  const int tid  = threadIdx.x;
  const int lane = tid & 31;
  const int wave = tid >> 5;
  const int token = blockIdx.x * 8 + wave;   // one wave32 per token

  __shared__ float sI[kE];
  __shared__ float sL[kE];
  if (tid < kE) { sI[tid] = 0.0f; sL[tid] = 0.0f; }
  __syncthreads();

  const float* xr = x + (size_t)token * kH;
  float logit[kE];
#pragma unroll
  for (int e = 0; e < kE; ++e) {
    float p = 0.0f;
    for (int h = lane; h < kH; h += 32) p += xr[h] * gw[e * kH + h];
#pragma unroll
    for (int off = 16; off > 0; off >>= 1) p += __shfl_xor(p, off, 32);
    logit[e] = p;   // all lanes hold full sum after xor reduction
  }

  // softmax over 8 experts
  float mx = logit[0];
#pragma unroll
  for (int e = 1; e < kE; ++e) mx = fmaxf(mx, logit[e]);
  float pe[kE];
  float s = 0.0f;
#pragma unroll
  for (int e = 0; e < kE; ++e) { pe[e] = __expf(logit[e] - mx); s += pe[e]; }
  const float inv = 1.0f / s;   // one precise div per token: fine
#pragma unroll
  for (int e = 0; e < kE; ++e) pe[e] *= inv;

  // top-2 (stable: earliest index wins ties, matching jax top_k)
  int i1 = 0;
#pragma unroll
  for (int e = 1; e < kE; ++e) if (pe[e] > pe[i1]) i1 = e;
  int i2 = (i1 == 0) ? 1 : 0;
#pragma unroll
  for (int e = 0; e < kE; ++e) if (e != i1 && pe[e] > pe[i2]) i2 = e;
  const float tinv = 1.0f / fmaxf(pe[i1] + pe[i2], 1e-8f);
  const float w1 = pe[i1] * tinv;
  const float w2 = pe[i2] * tinv;

  if (lane == 0) {
#pragma unroll
    for (int e = 0; e < kE; ++e) {
      const float we = (e == i1) ? w1 : ((e == i2) ? w2 : 0.0f);
      Wt[(size_t)token * kE + e] = we;
      atomicAdd(&sI[e], pe[e]);                              // LDS fp32 atomic
      if (e == i1 || e == i2) atomicAdd(&sL[e], 1.0f);
    }
  }
  __syncthreads();
  if (tid < kE) {
    unsafeAtomicAdd(&sums[tid], sI[tid]);
    unsafeAtomicAdd(&sums[kE + tid], sL[tid]);
  }
}

__global__ void finalize_kernel(const float* __restrict__ sums,
                                float* __restrict__ aux) {
  if (threadIdx.x == 0) {
    float a = 0.0f;
#pragma unroll
    for (int e = 0; e < kE; ++e) {
      const float imp  = sums[e] / (float)kT;
      const float load = sums[kE + e] / ((float)kT * 2.0f);
      a += (float)kE * imp * load;
    }
    *aux = a;
  }
}

// ---------------- fused expert FFN: (x@Wgu -> silu*u) @ Wdown -------------
__global__ __launch_bounds__(256, 1)
void moe_expert_kernel(const float* __restrict__ x,
                       const float* __restrict__ wgu,
                       const float* __restrict__ wdn,
                       const float* __restrict__ Wt,
                       float* __restrict__ out) {
  __shared__ float xs[kMT * kKSlice];   // 16 KB x K-slice
  __shared__ float hs[kMT * kChunk];    // 32 KB silu(g)*u chunk
  __shared__ float wr[kMT];
  __shared__ int s_any;

  const int tid  = threadIdx.x;
  const int lane = tid & 31;
  const int wave = tid >> 5;            // 0..7
  const int e    = blockIdx.y;
  const int token0 = blockIdx.x * kMT;

  if (tid == 0) s_any = 0;
  __syncthreads();
  if (tid < kMT) {
    const float w = Wt[(size_t)(token0 + tid) * kE + e];
    wr[tid] = w;
    if (w != 0.0f) atomicOr(&s_any, 1);
  }
  __syncthreads();
  if (!s_any) return;   // no token in this tile routed to expert e

  const float* wg = wgu + (size_t)e * kH * kNG;   // [1024, 5632] row-major
  const float* wd = wdn + (size_t)e * kI * kH;    // [2816, 1024] row-major

  const int klo = (lane >> 4) << 1;   // 0 (lanes 0-15) or 2 (lanes 16-31)
  const int ln  = lane & 15;
  const int rhi = (lane >> 4) << 3;   // C/D row offset: 0 or 8

  v8f acc[32];                        // y tile: 4 Msub x 8 Nsub per wave
#pragma unroll
  for (int i = 0; i < 32; ++i) acc[i] = (v8f)0.0f;

  for (int c = 0; c < kNChunk; ++c) {
    // ---------- phase 1: g,u = x_tile @ wg[:, chunk cols] ----------
    v8f ag[4], au[4];
#pragma unroll
    for (int m = 0; m < 4; ++m) { ag[m] = (v8f)0.0f; au[m] = (v8f)0.0f; }

    const int gcol = c * kChunk + wave * 16 + ln;
    const int ucol = gcol + kI;

    for (int kb = 0; kb < kH / kKSlice; ++kb) {
      for (int idx = tid; idx < (kMT * kKSlice / 4); idx += 256) {
        const int m = idx >> 4;            // token row 0..63
        const int q = (idx & 15) << 2;     // float offset in slice
        const float4 v =
            *(const float4*)(x + (size_t)(token0 + m) * kH + kb * kKSlice + q);
        *(float4*)(&xs[m * kKSlice + q]) = v;
      }
      __syncthreads();
      for (int ks = 0; ks < kKSlice; ks += 4) {
        const int krow = kb * kKSlice + ks + klo;
        v2f bg, bu;
        bg.x = wg[(size_t)krow * kNG + gcol];
        bg.y = wg[(size_t)(krow + 1) * kNG + gcol];
        bu.x = wg[(size_t)krow * kNG + ucol];
        bu.y = wg[(size_t)(krow + 1) * kNG + ucol];
#pragma unroll
        for (int m = 0; m < 4; ++m) {
          v2f a;
          const int row = m * 16 + ln;
          a.x = xs[row * kKSlice + ks + klo];
          a.y = xs[row * kKSlice + ks + klo + 1];
          ag[m] = wmma_f32_16x16x4(a, bg, ag[m]);
          au[m] = wmma_f32_16x16x4(a, bu, au[m]);
        }
      }
      __syncthreads();
    }

    // silu(g) * u -> hs
    const int cn = wave * 16 + ln;
#pragma unroll
    for (int m = 0; m < 4; ++m) {
#pragma unroll
      for (int r = 0; r < 8; ++r) {
        const int row = m * 16 + r + rhi;
        hs[row * kChunk + cn] = silu_mul(ag[m][r], au[m][r]);
      }
    }
    __syncthreads();

    // ---------- phase 2: y += h_chunk @ wd[chunk rows, :] ----------
    const int nb = wave * 128;
    for (int ks = 0; ks < kChunk; ks += 4) {
      v2f a[4];
#pragma unroll
      for (int m = 0; m < 4; ++m) {
        const int row = m * 16 + ln;
        a[m].x = hs[row * kChunk + ks + klo];
        a[m].y = hs[row * kChunk + ks + klo + 1];
      }
      const size_t krow = (size_t)(c * kChunk + ks + klo);
      const float* wr0 = wd + krow * kH;
#pragma unroll
      for (int j = 0; j < 8; ++j) {
        const int col = nb + j * 16 + ln;
        v2f b;
        b.x = wr0[col];
        b.y = wr0[kH + col];
#pragma unroll
        for (int m = 0; m < 4; ++m)
          acc[m * 8 + j] = wmma_f32_16x16x4(a[m], b, acc[m * 8 + j]);
      }
    }
    __syncthreads();  // hs reused next chunk
  }

  // ---------- scale by routing weight, accumulate into out ----------
#pragma unroll
  for (int m = 0; m < 4; ++m) {
#pragma unroll
    for (int r = 0; r < 8; ++r) {
      const int row = m * 16 + r + rhi;
      const float wv = wr[row];
      if (wv != 0.0f) {
        float* orow = out + (size_t)(token0 + row) * kH;
#pragma unroll
        for (int j = 0; j < 8; ++j) {
          const int col = wave * 128 + j * 16 + ln;
          unsafeAtomicAdd(orow + col, acc[m * 8 + j][r] * wv);
        }
      }
    }
  }
}

// ---------------- host launcher ----------------
extern "C" void kernel_launch(void* const* d_in, const int* in_sizes, int n_in,
                              void* d_out, int out_size, void* d_ws,
                              size_t ws_size, hipStream_t stream) {
  (void)in_sizes; (void)n_in; (void)ws_size;
  const float* x   = (const float*)d_in[0];
  const float* gw  = (const float*)d_in[1];
  const float* wgu = (const float*)d_in[2];
  const float* wdn = (const float*)d_in[3];
  float* out  = (float*)d_out;
  float* Wt   = (float*)d_ws;                  // [T, 8] dense combine weights
  float* sums = Wt + (size_t)kT * kE;          // [16] importance/load sums
  float* aux  = out + (size_t)kT * kH;         // aux_loss slot

  hipMemsetAsync(d_out, 0, (size_t)out_size * sizeof(float), stream);
  hipMemsetAsync(sums, 0, 2 * kE * sizeof(float), stream);

  router_kernel<<<dim3(kT / 8), dim3(256), 0, stream>>>(x, gw, Wt, sums);
  finalize_kernel<<<dim3(1), dim3(32), 0, stream>>>(sums, aux);
  moe_expert_kernel<<<dim3(kT / kMT, kE), dim3(256), 0, stream>>>(x, wgu, wdn,
                                                                  Wt, out);
}